// Encoder_66279935312283
// MI455X (gfx1250) — compile-verified
//
#include <hip/hip_runtime.h>

typedef float v2f __attribute__((ext_vector_type(2)));
typedef float v8f __attribute__((ext_vector_type(8)));

#define DIM 64
#define NGRAPHS 256
#define D3 192
#define BN_EPS 1e-5f

// ---------------------------------------------------------------------------
// Edge scatter-add: agg[dst] += h[src].  16 threads per edge, float4 chunks.
// Node features (12.8 MB) are L2-resident on MI455X (192 MB L2).
// ---------------------------------------------------------------------------
__global__ void scatter_add_kernel(const float* __restrict__ h,
                                   const int* __restrict__ src,
                                   const int* __restrict__ dst,
                                   float* __restrict__ agg, int nEdges) {
  int idx = blockIdx.x * blockDim.x + threadIdx.x;
  int e = idx >> 4;
  if (e >= nEdges) return;
  int q = (idx & 15) << 2;
  int s = src[e];
  int d = dst[e];
  const float4 v = *(const float4*)(h + (size_t)s * DIM + q);
  float* o = agg + (size_t)d * DIM + q;
  atomicAdd(o + 0, v.x);
  atomicAdd(o + 1, v.y);
  atomicAdd(o + 2, v.z);
  atomicAdd(o + 3, v.w);
}

// ---------------------------------------------------------------------------
// Pack weights into WMMA B-fragment pair layout:
//   Wp[(k/2)*N + c] = {W[k][c], W[k+1][c]}
// so the GEMM hot loop issues one b64 load per fragment instead of two b32.
// ---------------------------------------------------------------------------
__global__ void pack_w_kernel(const float* __restrict__ W,
                              float2* __restrict__ Wp, int K, int N) {
  int idx = blockIdx.x * blockDim.x + threadIdx.x;
  int total = (K >> 1) * N;
  if (idx >= total) return;
  int c = idx % N;
  int kp = idx / N;
  Wp[idx] = make_float2(W[(size_t)(2 * kp) * N + c],
                        W[(size_t)(2 * kp + 1) * N + c]);
}

// ---------------------------------------------------------------------------
// Fused GIN node MLP: z = relu(relu((h+agg)@W1+B1)@W2+B2), exact f32 via
// V_WMMA_F32_16X16X4_F32.  One wave = 16 node rows x 64 outputs; block = 8
// waves = 128 rows.  Intermediate 16x64 tile staged through wave-private LDS.
// Tail rows handled by clamp+mask so EXEC stays full (no per-iter saveexec).
// Also accumulates per-feature sum / sum-of-squares for BatchNorm.
// ---------------------------------------------------------------------------
__global__ __launch_bounds__(256) void gin_mlp_kernel(
    const float* __restrict__ h, const float* __restrict__ agg,
    const float2* __restrict__ Wp1, const float* __restrict__ B1,
    const float2* __restrict__ Wp2, const float* __restrict__ B2,
    float* __restrict__ z_out, float* __restrict__ bnsum,
    float* __restrict__ bnsq, int nNodes) {
  __shared__ float tlds[8][16 * DIM];
  __shared__ float s_sum[DIM];
  __shared__ float s_sq[DIM];

  const int wave = threadIdx.x >> 5;
  const int lane = threadIdx.x & 31;
  const int li = lane & 15;   // row (A) / col (B,C) index within tile
  const int hi = lane >> 4;   // selects K pair (A/B) or +8 row (C/D)
  const int rowBase = blockIdx.x * 128 + wave * 16;

  if (threadIdx.x < DIM) {
    s_sum[threadIdx.x] = 0.0f;
    s_sq[threadIdx.x] = 0.0f;
  }
  __syncthreads();

  // ---- GEMM1: T = (h+agg)[16x64] @ W1[64x64] ----
  const int r = rowBase + li;
  const int rc = min(r, nNodes - 1);           // always-valid load address
  const float rmask = (r < nNodes) ? 1.0f : 0.0f;
  const float* hrow = h + (size_t)rc * DIM;
  const float* arow = agg + (size_t)rc * DIM;

  v8f acc1[4] = {};
  for (int k0 = 0; k0 < DIM; k0 += 4) {
    const int kk = k0 + 2 * hi;
    v2f a;
    a.x = (hrow[kk] + arow[kk]) * rmask;
    a.y = (hrow[kk + 1] + arow[kk + 1]) * rmask;
    const float2* wrow = Wp1 + (size_t)(kk >> 1) * DIM + li;
#pragma unroll
    for (int nt = 0; nt < 4; ++nt) {
      float2 bb = wrow[nt * 16];
      v2f b;
      b.x = bb.x;
      b.y = bb.y;
      acc1[nt] = __builtin_amdgcn_wmma_f32_16x16x4_f32(
          false, a, false, b, (short)0, acc1[nt], false, false);
    }
  }

  // bias + relu, stage to wave-private LDS tile (C layout -> row major)
  float* t = &tlds[wave][0];
#pragma unroll
  for (int nt = 0; nt < 4; ++nt) {
#pragma unroll
    for (int v = 0; v < 8; ++v) {
      const int row = v + 8 * hi;
      const int col = nt * 16 + li;
      float val = acc1[nt][v] + B1[col];
      val = fmaxf(val, 0.0f);
      t[row * DIM + col] = val;
    }
  }

  // ---- GEMM2: Z = T[16x64] @ W2[64x64] ----  (tile is wave-private: no barrier)
  v8f acc2[4] = {};
  for (int k0 = 0; k0 < DIM; k0 += 4) {
    const int kk = k0 + 2 * hi;
    v2f a;
    a.x = t[li * DIM + kk];
    a.y = t[li * DIM + kk + 1];
    const float2* wrow = Wp2 + (size_t)(kk >> 1) * DIM + li;
#pragma unroll
    for (int nt = 0; nt < 4; ++nt) {
      float2 bb = wrow[nt * 16];
      v2f b;
      b.x = bb.x;
      b.y = bb.y;
      acc2[nt] = __builtin_amdgcn_wmma_f32_16x16x4_f32(
          false, a, false, b, (short)0, acc2[nt], false, false);
    }
  }

  // bias + relu, store, BN partial sums
#pragma unroll
  for (int nt = 0; nt < 4; ++nt) {
    const int col = nt * 16 + li;
    float s = 0.0f, sq = 0.0f;
#pragma unroll
    for (int v = 0; v < 8; ++v) {
      const int row = rowBase + v + 8 * hi;
      float val = acc2[nt][v] + B2[col];
      val = fmaxf(val, 0.0f);
      if (row < nNodes) {
        z_out[(size_t)row * DIM + col] = val;
        s += val;
        sq += val * val;
      }
    }
    // lanes L and L^16 hold the same column (rows v vs v+8): combine
    s += __shfl_xor(s, 16, 32);
    sq += __shfl_xor(sq, 16, 32);
    if (hi == 0) {
      atomicAdd(&s_sum[col], s);
      atomicAdd(&s_sq[col], sq);
    }
  }
  __syncthreads();
  if (threadIdx.x < DIM) {
    atomicAdd(&bnsum[threadIdx.x], s_sum[threadIdx.x]);
    atomicAdd(&bnsq[threadIdx.x], s_sq[threadIdx.x]);
  }
}

// ---------------------------------------------------------------------------
// BatchNorm apply (training-mode batch stats, biased variance), in place.
// ---------------------------------------------------------------------------
__global__ void bn_apply_kernel(float* __restrict__ z,
                                const float* __restrict__ bnsum,
                                const float* __restrict__ bnsq,
                                const float* __restrict__ gamma,
                                const float* __restrict__ beta, int nNodes) {
  int idx = blockIdx.x * blockDim.x + threadIdx.x;
  int total = nNodes * DIM;
  if (idx >= total) return;
  int col = idx & (DIM - 1);
  float invN = 1.0f / (float)nNodes;
  float mu = bnsum[col] * invN;
  float var = bnsq[col] * invN - mu * mu;
  float inv = rsqrtf(var + BN_EPS);
  z[idx] = (z[idx] - mu) * inv * gamma[col] + beta[col];
}

// ---------------------------------------------------------------------------
// Global-add-pool over sorted batch ids; register running sums flushed on
// graph-id change (few atomics per chunk instead of one per element).
// blockDim = (64, 4); each y-row handles 64 consecutive nodes, x = feature.
// ---------------------------------------------------------------------------
__global__ void pool_kernel(const float* __restrict__ z0,
                            const float* __restrict__ z1,
                            const float* __restrict__ z2,
                            const int* __restrict__ batch,
                            float* __restrict__ pooled, int nNodes) {
  const int d = threadIdx.x;
  const int seg = blockIdx.x * 4 + threadIdx.y;
  const int n0 = seg * 64;
  if (n0 >= nNodes) return;
  const int n1 = min(n0 + 64, nNodes);
  const float* zs[3] = {z0, z1, z2};
  float acc[3] = {0.0f, 0.0f, 0.0f};
  int curg = batch[n0];
  for (int n = n0; n < n1; ++n) {
    int g = batch[n];
    if (g != curg) {
#pragma unroll
      for (int l = 0; l < 3; ++l) {
        atomicAdd(&pooled[(size_t)curg * D3 + l * DIM + d], acc[l]);
        acc[l] = 0.0f;
      }
      curg = g;
    }
#pragma unroll
    for (int l = 0; l < 3; ++l) acc[l] += zs[l][(size_t)n * DIM + d];
  }
#pragma unroll
  for (int l = 0; l < 3; ++l)
    atomicAdd(&pooled[(size_t)curg * D3 + l * DIM + d], acc[l]);
}

// ---------------------------------------------------------------------------
// Readout dense layer: C = act(A[MxK] @ Wp[KxN packed] + B).  One wave per
// 16x16 tile, exact f32 WMMA.  M=256, N=K=192.
// ---------------------------------------------------------------------------
__global__ __launch_bounds__(128) void dense_kernel(
    const float* __restrict__ A, const float2* __restrict__ Wp,
    const float* __restrict__ B, float* __restrict__ C, int M, int N, int K,
    int doRelu) {
  const int wave = threadIdx.x >> 5;
  const int lane = threadIdx.x & 31;
  const int li = lane & 15;
  const int hi = lane >> 4;
  const int tilesN = N >> 4;
  const int tile = blockIdx.x * 4 + wave;
  if (tile >= (M >> 4) * tilesN) return;
  const int tm = tile / tilesN;
  const int tn = tile % tilesN;

  const float* arow = A + (size_t)(tm * 16 + li) * K;
  v8f acc = {};
  for (int k0 = 0; k0 < K; k0 += 4) {
    const int kk = k0 + 2 * hi;
    v2f a;
    a.x = arow[kk];
    a.y = arow[kk + 1];
    float2 bb = Wp[(size_t)(kk >> 1) * N + tn * 16 + li];
    v2f b;
    b.x = bb.x;
    b.y = bb.y;
    acc = __builtin_amdgcn_wmma_f32_16x16x4_f32(false, a, false, b, (short)0,
                                                acc, false, false);
  }
#pragma unroll
  for (int v = 0; v < 8; ++v) {
    const int row = tm * 16 + v + 8 * hi;
    const int col = tn * 16 + li;
    float val = acc[v] + B[col];
    if (doRelu) val = fmaxf(val, 0.0f);
    C[(size_t)row * N + col] = val;
  }
}

// ---------------------------------------------------------------------------
extern "C" void kernel_launch(void* const* d_in, const int* in_sizes, int n_in,
                              void* d_out, int out_size, void* d_ws,
                              size_t ws_size, hipStream_t stream) {
  const float* x = (const float*)d_in[0];
  const int* edge = (const int*)d_in[1];  // [2, E]
  const int* batch = (const int*)d_in[2];
  const float* W1 = (const float*)d_in[3];  // [3,64,64]
  const float* B1 = (const float*)d_in[4];  // [3,64]
  const float* W2 = (const float*)d_in[5];
  const float* B2 = (const float*)d_in[6];
  const float* gamma = (const float*)d_in[7];
  const float* beta = (const float*)d_in[8];
  const float* PW1 = (const float*)d_in[9];  // [192,192]
  const float* PB1 = (const float*)d_in[10];
  const float* PW2 = (const float*)d_in[11];
  const float* PB2 = (const float*)d_in[12];
  float* out = (float*)d_out;

  const int nNodes = in_sizes[0] / DIM;
  const int nEdges = in_sizes[1] / 2;
  const size_t featBytes = (size_t)nNodes * DIM * sizeof(float);

  char* ws = (char*)d_ws;
  float* agg = (float*)ws;        ws += featBytes;
  float* xs[3];
  for (int i = 0; i < 3; ++i) { xs[i] = (float*)ws; ws += featBytes; }
  float* bnsum = (float*)ws;      ws += DIM * sizeof(float);
  float* bnsq = (float*)ws;       ws += DIM * sizeof(float);
  float* pooled = (float*)ws;     ws += (size_t)NGRAPHS * D3 * sizeof(float);
  float* hidden = (float*)ws;     ws += (size_t)NGRAPHS * D3 * sizeof(float);
  float2* wp1 = (float2*)ws;      ws += 3 * (DIM / 2) * DIM * sizeof(float2);
  float2* wp2 = (float2*)ws;      ws += 3 * (DIM / 2) * DIM * sizeof(float2);
  float2* pwp1 = (float2*)ws;     ws += (D3 / 2) * D3 * sizeof(float2);
  float2* pwp2 = (float2*)ws;     ws += (D3 / 2) * D3 * sizeof(float2);

  const int* src = edge;
  const int* dst = edge + nEdges;

  // One-shot weight packing into WMMA fragment-pair layout.
  {
    const int wtot = (DIM / 2) * DIM;  // 2048 float2 per 64x64 matrix
    for (int l = 0; l < 3; ++l) {
      pack_w_kernel<<<(wtot + 255) / 256, 256, 0, stream>>>(
          W1 + (size_t)l * DIM * DIM, wp1 + (size_t)l * wtot, DIM, DIM);
      pack_w_kernel<<<(wtot + 255) / 256, 256, 0, stream>>>(
          W2 + (size_t)l * DIM * DIM, wp2 + (size_t)l * wtot, DIM, DIM);
    }
    const int ptot = (D3 / 2) * D3;  // 18432 float2 per 192x192 matrix
    pack_w_kernel<<<(ptot + 255) / 256, 256, 0, stream>>>(PW1, pwp1, D3, D3);
    pack_w_kernel<<<(ptot + 255) / 256, 256, 0, stream>>>(PW2, pwp2, D3, D3);
  }

  const float* h = x;
  for (int l = 0; l < 3; ++l) {
    hipMemsetAsync(agg, 0, featBytes, stream);
    hipMemsetAsync(bnsum, 0, 2 * DIM * sizeof(float), stream);

    {
      long long threads = (long long)nEdges * 16;
      int blocks = (int)((threads + 255) / 256);
      scatter_add_kernel<<<blocks, 256, 0, stream>>>(h, src, dst, agg, nEdges);
    }
    {
      int blocks = (nNodes + 127) / 128;
      gin_mlp_kernel<<<blocks, 256, 0, stream>>>(
          h, agg, wp1 + (size_t)l * (DIM / 2) * DIM, B1 + l * DIM,
          wp2 + (size_t)l * (DIM / 2) * DIM, B2 + l * DIM, xs[l], bnsum, bnsq,
          nNodes);
    }
    {
      int blocks = (nNodes * DIM + 255) / 256;
      bn_apply_kernel<<<blocks, 256, 0, stream>>>(xs[l], bnsum, bnsq,
                                                  gamma + l * DIM,
                                                  beta + l * DIM, nNodes);
    }
    h = xs[l];
  }

  hipMemsetAsync(pooled, 0, (size_t)NGRAPHS * D3 * sizeof(float), stream);
  {
    int blocks = (nNodes + 255) / 256;
    dim3 blk(64, 4);
    pool_kernel<<<blocks, blk, 0, stream>>>(xs[0], xs[1], xs[2], batch, pooled,
                                            nNodes);
  }

  // readout MLP: [256x192] @ [192x192] twice; 16*12 = 192 tiles, 4 waves/block
  dense_kernel<<<48, 128, 0, stream>>>(pooled, pwp1, PB1, hidden, NGRAPHS, D3,
                                       D3, 1);
  dense_kernel<<<48, 128, 0, stream>>>(hidden, pwp2, PB2, out, NGRAPHS, D3, D3,
                                       0);
}